// Moment_2774548873409
// MI455X (gfx1250) — compile-verified
//
#include <hip/hip_runtime.h>
#include <cstdint>
#include <cstddef>

typedef __attribute__((ext_vector_type(2))) float v2f;
typedef __attribute__((ext_vector_type(8))) float v8f;

#define LATENT 16
#define N2 136
#define N3 816
#define N4 3876
#define NTAB (N2 + N3 + N4)   // 4828, divisible by 4
#define O2 0
#define O3 N2
#define O4 (N2 + N3)
#define T3N 9    // ceil(136/16) tiles at order 3
#define T4N 51   // 816/16 tiles at order 4
#define CH_TOTAL 4845
#define OFF1 1
#define OFF2 17
#define OFF3 153
#define OFF4 969
#define WAVES_PER_BLOCK 8

struct Tables {
  uint8_t tab[NTAB];   // tile-local index: (k*16 + j) & 255, per level, concatenated
  int ts3[T3N + 1];    // first pair index per 16-row tile (level 3)
  int ts4[T4N + 1];    // first pair index per 16-row tile (level 4)
  int n2, n3, n4;
};

// Exactly replicates _build_indices(16, 4): suffix sums, offsets = n - S,
// keep[k, j] = k >= offsets[j], flattened row-major (k major, j minor).
constexpr Tables build_tables() {
  Tables t = {};
  long c[LATENT] = {};
  for (int i = 0; i < LATENT; ++i) c[i] = 1;
  long S[LATENT] = {};
  int flat3[N3] = {};
  int flat4[N4] = {};
  {  // pass 1 -> level-2 table (single tile, flat < 256)
    long s = 0;
    for (int i = LATENT - 1; i >= 0; --i) { s += c[i]; S[i] = s; }
    int n = (int)S[0], idx = 0;
    for (int k = 0; k < n; ++k)
      for (int j = 0; j < LATENT; ++j)
        if (k >= n - (int)S[j]) t.tab[O2 + idx++] = (uint8_t)(k * LATENT + j);
    t.n2 = idx;
    for (int i = 0; i < LATENT; ++i) c[i] = S[i];
  }
  {  // pass 2 -> level-3 table
    long s = 0;
    for (int i = LATENT - 1; i >= 0; --i) { s += c[i]; S[i] = s; }
    int n = (int)S[0], idx = 0;
    for (int k = 0; k < n; ++k)
      for (int j = 0; j < LATENT; ++j)
        if (k >= n - (int)S[j]) {
          flat3[idx] = k * LATENT + j;
          t.tab[O3 + idx] = (uint8_t)(flat3[idx] & 255);
          ++idx;
        }
    t.n3 = idx;
    for (int i = 0; i < LATENT; ++i) c[i] = S[i];
  }
  {  // pass 3 -> level-4 table
    long s = 0;
    for (int i = LATENT - 1; i >= 0; --i) { s += c[i]; S[i] = s; }
    int n = (int)S[0], idx = 0;
    for (int k = 0; k < n; ++k)
      for (int j = 0; j < LATENT; ++j)
        if (k >= n - (int)S[j]) {
          flat4[idx] = k * LATENT + j;
          t.tab[O4 + idx] = (uint8_t)(flat4[idx] & 255);
          ++idx;
        }
    t.n4 = idx;
  }
  {  // tile start indices (pairs are k-major sorted -> flat sorted ascending)
    int i3 = 0;
    for (int tt = 0; tt <= T3N; ++tt) {
      while (i3 < N3 && (flat3[i3] >> 8) < tt) ++i3;
      t.ts3[tt] = i3;
    }
    int i4 = 0;
    for (int tt = 0; tt <= T4N; ++tt) {
      while (i4 < N4 && (flat4[i4] >> 8) < tt) ++i4;
      t.ts4[tt] = i4;
    }
  }
  return t;
}

constexpr Tables h_tables = build_tables();
static_assert(h_tables.n2 == N2, "level-2 pair count mismatch");
static_assert(h_tables.n3 == N3, "level-3 pair count mismatch");
static_assert(h_tables.n4 == N4, "level-4 pair count mismatch");
static_assert(h_tables.ts3[T3N] == N3, "ts3 mismatch");
static_assert(h_tables.ts4[T4N] == N4, "ts4 mismatch");
static_assert((NTAB & 3) == 0, "table not dword-divisible");

__constant__ Tables g_tables = build_tables();

// Scatter a 16x16 f32 WMMA C/D accumulator to LDS in row-major (M*16+N) order.
// C/D layout: VGPR r, lanes 0-15 -> M=r, N=lane; lanes 16-31 -> M=r+8, N=lane-16.
__device__ __forceinline__ void store_tile(float* tile, int lane, v8f d) {
  const int col = lane & 15;
  const int hi = (lane >> 4) << 3;
#pragma unroll
  for (int r = 0; r < 8; ++r)
    tile[(r + hi) * 16 + col] = d[r];
}

// Gather tile-local indexed values for output range [beg, end); 2-wide unrolled
// so LDS index/data loads batch under one wait instead of draining per element.
template <bool KEEP_FEATS>
__device__ __forceinline__ void gather_store(const uint8_t* __restrict__ tab,
                                             const float* __restrict__ tile,
                                             float* __restrict__ fdst,
                                             float* __restrict__ gdst,
                                             int beg, int end, int lane) {
  int p = beg + lane;
  for (; p + 32 < end; p += 64) {
    const int i0 = tab[p];
    const int i1 = tab[p + 32];
    const float v0 = tile[i0];
    const float v1 = tile[i1];
    if (KEEP_FEATS) { fdst[p] = v0; fdst[p + 32] = v1; }
    __builtin_nontemporal_store(v0, gdst + p);
    __builtin_nontemporal_store(v1, gdst + p + 32);
  }
  if (p < end) {
    const float v = tile[tab[p]];
    if (KEEP_FEATS) fdst[p] = v;
    __builtin_nontemporal_store(v, gdst + p);
  }
}

// One wave (wave32) per (b,t) position; 8 waves per 256-thread block.
__global__ __launch_bounds__(256) void moment_wmma_kernel(
    const float* __restrict__ x, float* __restrict__ out, int npos) {
  __shared__ uint8_t s_tab[NTAB];                  // 4.8 KB shared index table
  __shared__ float s_f1[WAVES_PER_BLOCK][N2];      // order-2 feats per wave
  __shared__ float s_f2[WAVES_PER_BLOCK][N3];      // order-3 feats per wave
  __shared__ float s_tile[WAVES_PER_BLOCK][256];   // 16x16 outer-product tile per wave

  const int tid = threadIdx.x;
  {  // stage index table as dwords
    const uint32_t* src = (const uint32_t*)g_tables.tab;
    uint32_t* dst = (uint32_t*)s_tab;
    for (int i = tid; i < NTAB / 4; i += 256) dst[i] = src[i];
  }
  __syncthreads();

  const int lane = tid & 31;
  const int wave = tid >> 5;
  const int pos = blockIdx.x * WAVES_PER_BLOCK + wave;
  if (pos >= npos) return;  // wave-uniform exit; survivors keep EXEC all-1s

  const float* xp = x + (size_t)pos * LATENT;
  float* op = out + (size_t)pos * CH_TOTAL;

  const float xv = (lane < LATENT) ? xp[lane] : 0.0f;
  if (lane == 0) __builtin_nontemporal_store(1.0f, op);                  // order 0
  if (lane < LATENT) __builtin_nontemporal_store(xv, op + OFF1 + lane);  // order 1

  // A (16x4 f32): VGPR0 lanes 0-15 = K=0 column; all else zero.
  // B (4x16 f32): VGPR0 lanes 0-15 = K=0 row; all else zero.  Same lane pattern.
  v2f acol; acol.x = xv; acol.y = 0.0f;
  const v2f brow = acol;
  const v8f czero = {0.f, 0.f, 0.f, 0.f, 0.f, 0.f, 0.f, 0.f};

  float* tile = s_tile[wave];
  float* f1 = s_f1[wave];
  float* f2 = s_f2[wave];

  // ---- order 2: one outer-product tile x . x^T ----
  v8f d2 = __builtin_amdgcn_wmma_f32_16x16x4_f32(
      false, acol, false, brow, (short)0, czero, false, false);
  store_tile(tile, lane, d2);
  gather_store<true>(s_tab + O2, tile, f1, op + OFF2, 0, N2, lane);

  // ---- order 3: 9 tiles of feats1-chunk . x^T ----
  for (int t = 0; t < T3N; ++t) {
    const int row = t * 16 + lane;
    const float pv = (lane < 16 && row < N2) ? f1[row] : 0.0f;
    v2f a; a.x = pv; a.y = 0.0f;
    v8f d3 = __builtin_amdgcn_wmma_f32_16x16x4_f32(
        false, a, false, brow, (short)0, czero, false, false);
    store_tile(tile, lane, d3);
    gather_store<true>(s_tab + O3, tile, f2, op + OFF3,
                       g_tables.ts3[t], g_tables.ts3[t + 1], lane);
  }

  // ---- order 4: 51 tiles of feats2-chunk . x^T ----
  for (int t = 0; t < T4N; ++t) {
    const float pv = (lane < 16) ? f2[t * 16 + lane] : 0.0f;  // 51*16 == 816
    v2f a; a.x = pv; a.y = 0.0f;
    v8f d4 = __builtin_amdgcn_wmma_f32_16x16x4_f32(
        false, a, false, brow, (short)0, czero, false, false);
    store_tile(tile, lane, d4);
    gather_store<false>(s_tab + O4, tile, nullptr, op + OFF4,
                        g_tables.ts4[t], g_tables.ts4[t + 1], lane);
  }
}

extern "C" void kernel_launch(void* const* d_in, const int* in_sizes, int n_in,
                              void* d_out, int out_size, void* d_ws, size_t ws_size,
                              hipStream_t stream) {
  const float* x = (const float*)d_in[0];
  float* out = (float*)d_out;
  const int npos = in_sizes[0] / LATENT;                         // B*T = 8192
  const int blocks = (npos + WAVES_PER_BLOCK - 1) / WAVES_PER_BLOCK;
  moment_wmma_kernel<<<blocks, 256, 0, stream>>>(x, out, npos);
}